// Rasterize_32289564131765
// MI455X (gfx1250) — compile-verified
//
#include <hip/hip_runtime.h>

typedef float v2f __attribute__((ext_vector_type(2)));
typedef float v8f __attribute__((ext_vector_type(8)));

#define IMG    256
#define NEARP  0.1f
#define FARP   100.0f
#define EPS_T  0.001f
#define BANDPX 16                    // 16x16-pixel bands
#define NBANDS (2 * 16 * 16)         // 512 (B=2)
#define CAP    1024                  // max faces per band (mean ~75 after cull)

__device__ __forceinline__ float px_x(int c) {
    return (2.0f * (float)c + 1.0f - (float)IMG) / (float)IMG;
}
__device__ __forceinline__ float px_y(int r) {
    return -((2.0f * (float)r + 1.0f - (float)IMG) / (float)IMG);
}

// ---------------------------------------------------------------------------
// Deterministic 2D binning (one block per band, ballot-scan in face order).
// Culls faces that cannot win: bbox outside the band (inclusive compares,
// identical xp/yp formulas as the raster kernel -> exact superset of hits)
// OR det <= 0 (back-facing / degenerate: the reference's inside test
// (all u_i >= 0 with sum u_i = det) is unsatisfiable for det < 0, and
// det == 0 yields zp = 1e10 > far).  Emits, in face order:
//   - 8 x float2 coefficient entries in WMMA A-operand lane layout:
//       row m = 4*slot + comp:  lane m -> (a,b),  lane m+16 -> (c,0)
//     where u_i(p) = a_i*xp + b_i*yp + c_i  (u_i = w_i*det) and
//     row3 = sum_i row_i/z_i so q(p) = det * inv_zp.
//   - the face index, same order.
// Tail padded to a multiple of 4 with zero entries (q=0 => never valid).
// ---------------------------------------------------------------------------
__global__ void __launch_bounds__(256)
bin_faces(const float* __restrict__ faces, float2* __restrict__ coeff,
          int* __restrict__ idxArr, int* __restrict__ counts, int F)
{
    const int blk = blockIdx.x;              // b*256 + by*16 + bx
    const int b  = blk >> 8;
    const int by = (blk >> 4) & 15;
    const int bx = blk & 15;

    const float ypTop = px_y(by * BANDPX);
    const float ypBot = px_y(by * BANDPX + BANDPX - 1);
    const float xpLo  = px_x(bx * BANDPX);
    const float xpHi  = px_x(bx * BANDPX + BANDPX - 1);

    const size_t bandC = (size_t)blk * (CAP * 8);  // float2 units
    const size_t bandI = (size_t)blk * CAP;

    __shared__ int waveSums[8];
    const int tid = threadIdx.x, waveId = tid >> 5, lane = tid & 31;
    int outBase = 0;

    for (int chunk = 0; chunk < F; chunk += 256) {
        const int f = chunk + tid;
        bool pred = false;
        float x0 = 0, y0 = 0, z0 = 0, x1 = 0, y1 = 0, z1 = 0, x2 = 0, y2 = 0, z2 = 0;
        float c0 = 0, c1 = 0, c2 = 0;
        if (f < F) {
            const float* fp = faces + (size_t)(b * F + f) * 9;
            x0 = fp[0]; y0 = fp[1]; z0 = fp[2];
            x1 = fp[3]; y1 = fp[4]; z1 = fp[5];
            x2 = fp[6]; y2 = fp[7]; z2 = fp[8];
            float xmn = fminf(x0, fminf(x1, x2)), xmx = fmaxf(x0, fmaxf(x1, x2));
            float ymn = fminf(y0, fminf(y1, y2)), ymx = fmaxf(y0, fmaxf(y1, y2));
            c0 = x1 * y2 - x2 * y1;
            c1 = x2 * y0 - x0 * y2;
            c2 = x0 * y1 - x1 * y0;
            float det = c0 + c1 + c2;
            pred = (xmn <= xpHi) & (xmx >= xpLo) & (ymn <= ypTop) & (ymx >= ypBot)
                 & (det > 0.0f);
        }
        unsigned mask = (unsigned)__ballot(pred);
        int prefix = __popc(mask & ((1u << lane) - 1u));
        if (lane == 0) waveSums[waveId] = __popc(mask);
        __syncthreads();
        int woff = 0, ctot = 0;
        #pragma unroll
        for (int w = 0; w < 8; ++w) {
            int s = waveSums[w];
            if (w < waveId) woff += s;
            ctot += s;
        }
        const int pos = outBase + woff + prefix;
        if (pred && pos < CAP) {
            float a[4], bb[4], c[4];
            a[0] = y1 - y2; bb[0] = x2 - x1; c[0] = c0;
            a[1] = y2 - y0; bb[1] = x0 - x2; c[1] = c1;
            a[2] = y0 - y1; bb[2] = x1 - x0; c[2] = c2;
            float r0 = 1.0f / z0, r1 = 1.0f / z1, r2 = 1.0f / z2;
            a[3]  = a[0] * r0 + a[1] * r1 + a[2] * r2;
            bb[3] = bb[0] * r0 + bb[1] * r1 + bb[2] * r2;
            c[3]  = c[0] * r0 + c[1] * r1 + c[2] * r2;
            const size_t e0 = bandC + (size_t)(pos >> 2) * 32 + (pos & 3) * 4;
            #pragma unroll
            for (int k = 0; k < 4; ++k) {
                coeff[e0 + k]      = make_float2(a[k], bb[k]);
                coeff[e0 + 16 + k] = make_float2(c[k], 0.0f);
            }
            idxArr[bandI + pos] = f;
        }
        outBase += ctot;
        __syncthreads();
    }

    const int cnt = min(outBase, CAP);
    const int padded = (cnt + 3) & ~3;
    if (tid < padded - cnt) {                      // zero the pad slots
        const int pos = cnt + tid;
        const size_t e0 = bandC + (size_t)(pos >> 2) * 32 + (pos & 3) * 4;
        #pragma unroll
        for (int k = 0; k < 4; ++k) {
            coeff[e0 + k]      = make_float2(0.0f, 0.0f);
            coeff[e0 + 16 + k] = make_float2(0.0f, 0.0f);
        }
        idxArr[bandI + pos] = -1;
    }
    if (tid == 0) counts[blk] = cnt;
}

// ---------------------------------------------------------------------------
// One wave32 per 16-pixel row segment. Scalar (readfirstlane) band id and
// trip count -> SGPR loop. Software-pipelined A-tile loads feed
// V_WMMA_F32_16X16X4_F32 (16 pixels x 4 faces per op). Miss test is just
// three compares (det>0 guaranteed by binning); Sum(u), near/far, the
// division-free z-race (det_a*q_b < det_b*q_a) and the index load sit
// behind a wave-uniform __any() branch.
// ---------------------------------------------------------------------------
__global__ void __launch_bounds__(256)
raster_wmma(const float* __restrict__ faces, const float* __restrict__ tex,
            const float2* __restrict__ coeff, const int* __restrict__ idxArr,
            const int* __restrict__ counts, float* __restrict__ out, int F)
{
    const int lane  = threadIdx.x & 31;
    const int gwave = blockIdx.x * (blockDim.x >> 5) + (threadIdx.x >> 5);
    const int tilesPerBatch = (IMG * IMG) / 16;          // 4096
    const int b    = gwave / tilesPerBatch;
    const int tile = gwave % tilesPerBatch;
    const int row  = tile >> 4;
    const int tcol = tile & 15;
    const int col  = (tcol << 4) + (lane & 15);

    const float xp = px_x(col);
    const float yp = px_y(row);

    // B-matrix (4x16): VGPR0 = K0|K2 -> (xp|1), VGPR1 = K1|K3 -> (yp|0)
    v2f bv;
    bv[0] = (lane < 16) ? xp : 1.0f;
    bv[1] = (lane < 16) ? yp : 0.0f;

    const int bandId = __builtin_amdgcn_readfirstlane(b * 256 + (row >> 4) * 16 + tcol);
    const int cnt = __builtin_amdgcn_readfirstlane(counts[bandId]);
    const int nT  = (cnt + 3) >> 2;                      // pad entries are zero
    const float2* cB = coeff + (size_t)bandId * (CAP * 8);
    const int*    iB = idxArr + (size_t)bandId * CAP;

    __builtin_prefetch(cB, 0, 1);                        // global_prefetch_b8

    float bd = FARP, bq = 1.0f;                 // best zp = bd/bq (init = FAR)
    float bu0 = 0.0f, bu1 = 0.0f, bu2 = 0.0f;
    int   bface = -1;
    const int fHalf = (lane < 16) ? 0 : 2;

    if (nT > 0) {
        v2f av = *(const v2f*)&cB[(size_t)lane];
        for (int ft = 0; ft < nT; ++ft) {
            // prefetch next tile (workspace has +256B slack past the array)
            v2f avn = *(const v2f*)&cB[(size_t)(ft + 1) * 32 + lane];
            v8f cz = {};
            v8f d  = __builtin_amdgcn_wmma_f32_16x16x4_f32(
                         false, av, false, bv, (short)0, cz, false, false);

            float u0a = d[0], u1a = d[1], u2a = d[2], qa = d[3];
            float u0b = d[4], u1b = d[5], u2b = d[6], qb = d[7];
            bool ina = (u0a >= 0.0f) & (u1a >= 0.0f) & (u2a >= 0.0f);
            bool inb = (u0b >= 0.0f) & (u1b >= 0.0f) & (u2b >= 0.0f);

            if (__any(ina | inb)) {              // wave-uniform skip
                int2 fid = *(const int2*)&iB[ft * 4 + fHalf];
                float da = u0a + u1a + u2a;      // per-pixel det (matches ref)
                float db = u0b + u1b + u2b;
                bool ua = ina & (da > NEARP * qa) & (da < FARP * qa)
                              & (da * bq < bd * qa);
                bd  = ua ? da  : bd;   bq  = ua ? qa  : bq;
                bu0 = ua ? u0a : bu0;  bu1 = ua ? u1a : bu1;  bu2 = ua ? u2a : bu2;
                bface = ua ? fid.x : bface;
                bool ub = inb & (db > NEARP * qb) & (db < FARP * qb)
                              & (db * bq < bd * qb);
                bd  = ub ? db  : bd;   bq  = ub ? qb  : bq;
                bu0 = ub ? u0b : bu0;  bu1 = ub ? u1b : bu1;  bu2 = ub ? u2b : bu2;
                bface = ub ? fid.y : bface;
            }
            av = avn;
        }
    }

    // Merge half-waves (lane <-> lane^16); lower half owns earlier faces,
    // strict < keeps "first minimum" semantics.
    float od  = __shfl_xor(bd,   16, 32);
    float oq  = __shfl_xor(bq,   16, 32);
    float ou0 = __shfl_xor(bu0,  16, 32);
    float ou1 = __shfl_xor(bu1,  16, 32);
    float ou2 = __shfl_xor(bu2,  16, 32);
    int   of  = __shfl_xor(bface, 16, 32);
    bool take = (od * bq < bd * oq);
    float md  = take ? od  : bd;
    float mq  = take ? oq  : bq;
    float mu0 = take ? ou0 : bu0;
    float mu1 = take ? ou1 : bu1;
    float mu2 = take ? ou2 : bu2;
    int   mf  = take ? of  : bface;

    if (lane < 16) {
        const size_t P   = (size_t)IMG * IMG;
        const size_t pix = (size_t)b * P + (size_t)row * IMG + col;
        float r = 0.0f, g = 0.0f, bl = 0.0f, alpha = 0.0f, zout = FARP;

        if (mf >= 0) {
            float zp = md / mq;
            float invdet = 1.0f / md;
            float w0 = fminf(fmaxf(mu0 * invdet, 0.0f), 1.0f);
            float w1 = fminf(fmaxf(mu1 * invdet, 0.0f), 1.0f);
            float w2 = fminf(fmaxf(mu2 * invdet, 0.0f), 1.0f);
            float is = 1.0f / fmaxf(w0 + w1 + w2, 1e-10f);
            w0 *= is; w1 *= is; w2 *= is;

            const float* fp = faces + (size_t)(b * F + mf) * 9;
            float t0 = fminf(fmaxf(w0 * 3.0f * zp / fp[2], 0.0f), 3.0f - EPS_T);
            float t1 = fminf(fmaxf(w1 * 3.0f * zp / fp[5], 0.0f), 3.0f - EPS_T);
            float t2 = fminf(fmaxf(w2 * 3.0f * zp / fp[8], 0.0f), 3.0f - EPS_T);
            int   l0 = (int)t0, l1 = (int)t1, l2 = (int)t2;
            float fr0 = t0 - (float)l0, fr1 = t1 - (float)l1, fr2 = t2 - (float)l2;

            const float* tb = tex + (size_t)(b * F + mf) * 64 * 3;
            #pragma unroll
            for (int pn = 0; pn < 8; ++pn) {
                int cc0 = pn & 1, cc1 = (pn >> 1) & 1, cc2 = (pn >> 2) & 1;
                float wc = (cc0 ? fr0 : 1.0f - fr0) *
                           (cc1 ? fr1 : 1.0f - fr1) *
                           (cc2 ? fr2 : 1.0f - fr2);
                int lin = ((l0 + cc0) * 4 + (l1 + cc1)) * 4 + (l2 + cc2);
                const float* tp = tb + (size_t)lin * 3;
                r += wc * tp[0]; g += wc * tp[1]; bl += wc * tp[2];
            }
            alpha = 1.0f;
            zout  = zp;
        }

        float* rgbOut = out;                       // (B,256,256,3)
        float* aOut   = out + 2 * P * 3;           // (B,256,256)
        float* zOut   = aOut + 2 * P;              // (B,256,256)
        rgbOut[pix * 3 + 0] = r;
        rgbOut[pix * 3 + 1] = g;
        rgbOut[pix * 3 + 2] = bl;
        aOut[pix] = alpha;
        zOut[pix] = zout;
    }
}

extern "C" void kernel_launch(void* const* d_in, const int* in_sizes, int n_in,
                              void* d_out, int out_size, void* d_ws, size_t ws_size,
                              hipStream_t stream)
{
    const float* faces = (const float*)d_in[0];      // (2, F, 3, 3) f32
    const float* tex   = (const float*)d_in[1];      // (2, F, 4,4,4, 3) f32
    const int F = in_sizes[0] / 18;                  // B=2, 9 floats per face

    // workspace layout (~34 MB + 256B prefetch slack):
    //   counts (1024 ints) | face-index lists (512*CAP) | A-tiles (512*CAP*8 f2)
    int*    counts = (int*)d_ws;
    int*    idxArr = counts + 1024;
    float2* coeff  = (float2*)(idxArr + (size_t)NBANDS * CAP);

    bin_faces<<<NBANDS, 256, 0, stream>>>(faces, coeff, idxArr, counts, F);

    const int nWaves  = 2 * (IMG * IMG / 16);        // 8192 waves, 1 per pixel tile
    const int threads = nWaves * 32;
    raster_wmma<<<threads / 256, 256, 0, stream>>>(faces, tex, coeff, idxArr,
                                                   counts, (float*)d_out, F);
}